// MCMHEdgeDecoder_69681549410500
// MI455X (gfx1250) — compile-verified
//
#include <hip/hip_runtime.h>
#include <hip/hip_bf16.h>
#include <stdint.h>

typedef __attribute__((ext_vector_type(2))) float v2f;
typedef __attribute__((ext_vector_type(8))) float v8f;

#define D_FEAT 128

// ---------------------------------------------------------------------------
// Phase 1: per-node scores s1[n] = W1 . X[n],  s2[n] = W2 . X[n]
// One wave (32 lanes) computes a 16-node tile using V_WMMA_F32_16X16X4_F32.
//   A (16x4, f32): lane L<16 holds row M=L, K={k,k+1}; lane L>=16 holds
//                  row M=L-16, K={k+2,k+3}  (ISA 7.12.2, 32-bit A 16x4).
//   B (4x16, f32): column 0 = W1 chunk, column 1 = W2 chunk, rest zero.
//   D (16x16, f32): element (M,N) -> lane N (N<16: VGPR M%8, half-wave by M/8).
// EXEC is kept all-1s around every WMMA (mask via arithmetic, not branches).
// ---------------------------------------------------------------------------
__global__ void __launch_bounds__(256)
node_scores_wmma(const float* __restrict__ X,
                 const float* __restrict__ W1,
                 const float* __restrict__ W2,
                 float* __restrict__ s1,
                 float* __restrict__ s2,
                 int ntiles)
{
    const int lane = threadIdx.x & 31;
    const int wave = threadIdx.x >> 5;
    const int tile = blockIdx.x * (blockDim.x >> 5) + wave;
    if (tile >= ntiles) return;                    // wave-uniform exit

    const int n   = lane & 15;                     // N index (B/C/D column)
    const int kho = (lane >> 4) * 2;               // K sub-offset: 0 or 2
    const int row = (tile << 4) + n;               // this lane's A-row M

    const float* __restrict__ xrow = X + (size_t)row * D_FEAT + kho;
    // Branchless B construction: lanes n==1 read W2, others read W1; a mask
    // zeroes columns >= 2 so EXEC never diverges across a WMMA.
    const float* __restrict__ wp = (n == 1) ? W2 : W1;
    const float  colmask = (n < 2) ? 1.0f : 0.0f;

    v8f acc = {};
#pragma unroll 8
    for (int k = 0; k < D_FEAT; k += 4) {
        v2f a;
        a.x = xrow[k + 0];
        a.y = xrow[k + 1];
        v2f b;
        b.x = wp[k + kho + 0] * colmask;
        b.y = wp[k + kho + 1] * colmask;
        acc = __builtin_amdgcn_wmma_f32_16x16x4_f32(
            /*neg_a=*/false, a, /*neg_b=*/false, b,
            /*c_mod=*/(short)0, acc, /*reuse_a=*/false, /*reuse_b=*/false);
    }

    // D layout: VGPR r, lanes 0-15 -> (M=r, N=lane); lanes 16-31 -> (M=8+r).
    // Column N==0 is s1, column N==1 is s2.
    const int mbase = (tile << 4) + ((lane >> 4) << 3);
    if (n < 2) {
        float* __restrict__ op = (n == 0) ? s1 : s2;
#pragma unroll
        for (int r = 0; r < 8; ++r)
            op[mbase + r] = acc[r];
    }
}

// Scalar fallback for a node-count tail that is not a multiple of 16.
__global__ void node_scores_tail(const float* __restrict__ X,
                                 const float* __restrict__ W1,
                                 const float* __restrict__ W2,
                                 float* __restrict__ s1,
                                 float* __restrict__ s2,
                                 int first, int n_nodes)
{
    int node = first + blockIdx.x * blockDim.x + threadIdx.x;
    if (node >= n_nodes) return;
    const float* xr = X + (size_t)node * D_FEAT;
    float a1 = 0.f, a2 = 0.f;
#pragma unroll 8
    for (int k = 0; k < D_FEAT; ++k) {
        float x = xr[k];
        a1 = fmaf(x, W1[k], a1);
        a2 = fmaf(x, W2[k], a2);
    }
    s1[node] = a1;
    s2[node] = a2;
}

// ---------------------------------------------------------------------------
// Phase 2: out[e] = s1[src[e]] + s2[dst[e]]
// Streams 16 B of int64 indices per edge; the two 4-B gathers hit the
// 400 KB score arrays, which are L2-resident (192 MB L2).
// ---------------------------------------------------------------------------
__global__ void __launch_bounds__(256)
edge_combine(const long long* __restrict__ edge_index,
             const float* __restrict__ s1,
             const float* __restrict__ s2,
             float* __restrict__ out,
             int n_edges)
{
    int e = blockIdx.x * blockDim.x + threadIdx.x;
    if (e >= n_edges) return;
    long long s = edge_index[e];                       // row 0: src
    long long d = edge_index[(size_t)n_edges + e];     // row 1: dst
    out[e] = s1[s] + s2[d];
}

extern "C" void kernel_launch(void* const* d_in, const int* in_sizes, int n_in,
                              void* d_out, int out_size, void* d_ws, size_t ws_size,
                              hipStream_t stream)
{
    const float*     X  = (const float*)d_in[0];
    const long long* EI = (const long long*)d_in[1];   // int64 [2, E]
    const float*     W1 = (const float*)d_in[2];
    const float*     W2 = (const float*)d_in[3];
    float*           out = (float*)d_out;

    const int n_nodes = in_sizes[0] / D_FEAT;
    const int n_edges = in_sizes[1] / 2;

    float* s1 = (float*)d_ws;
    float* s2 = s1 + n_nodes;

    // Phase 1: WMMA node scores (16 nodes per wave, 8 waves per block).
    const int ntiles = n_nodes >> 4;
    if (ntiles > 0) {
        const int waves_per_block = 256 / 32;
        const int nblocks = (ntiles + waves_per_block - 1) / waves_per_block;
        node_scores_wmma<<<nblocks, 256, 0, stream>>>(X, W1, W2, s1, s2, ntiles);
    }
    const int tail_first = ntiles << 4;
    if (tail_first < n_nodes) {
        const int rem = n_nodes - tail_first;
        node_scores_tail<<<(rem + 255) / 256, 256, 0, stream>>>(
            X, W1, W2, s1, s2, tail_first, n_nodes);
    }

    // Phase 2: gather-and-add over edges.
    edge_combine<<<(n_edges + 255) / 256, 256, 0, stream>>>(EI, s1, s2, out, n_edges);

    (void)n_in; (void)out_size; (void)ws_size;
}